// HybridSampler_65481071406452
// MI455X (gfx1250) — compile-verified
//
#include <hip/hip_runtime.h>

typedef float v2f __attribute__((ext_vector_type(2)));
typedef float v8f __attribute__((ext_vector_type(8)));

// branchless tanh: sign(x) * (1 - e^{-2|x|}) / (1 + e^{-2|x|})
// e^{-2|x|} = 2^{|x| * (-2*log2(e))}  -> one v_mul + one v_exp_f32
__device__ __forceinline__ float fast_tanh(float x) {
  const float ax = __builtin_fabsf(x);
  const float t  = __builtin_amdgcn_exp2f(ax * -2.8853900817779268f);
  const float r  = (1.0f - t) * __builtin_amdgcn_rcpf(1.0f + t);
  return __builtin_copysignf(r, x);                  // v_bfi
}

// sin/cos of th/2 via native v_sin/v_cos (input scaled by 1/(2*pi), shared mul)
#define SCHALF(th, ss, cc)                                                   \
  const float u_##ss = (th) * 0.07957747154594767f; /* 0.5/(2*pi) */         \
  const float ss = __builtin_amdgcn_sinf(u_##ss);                            \
  const float cc = __builtin_amdgcn_cosf(u_##ss)

// RY on qubit0: pairs (s0,s2) and (s1,s3)
#define RY_Q0(th) do { SCHALF(th, ss, cc);                                   \
    const float a0=s0, a1=s1, b0=s2, b1v=s3;                                 \
    s0 = cc*a0 - ss*b0;  s1 = cc*a1 - ss*b1v;                                \
    s2 = ss*a0 + cc*b0;  s3 = ss*a1 + cc*b1v; } while (0)

// RY on qubit1: pairs (s0,s1) and (s2,s3)
#define RY_Q1(th) do { SCHALF(th, ss, cc);                                   \
    const float a0=s0, b0=s1, a1=s2, b1v=s3;                                 \
    s0 = cc*a0 - ss*b0;  s1 = ss*a0 + cc*b0;                                 \
    s2 = cc*a1 - ss*b1v; s3 = ss*a1 + cc*b1v; } while (0)

__global__ __launch_bounds__(256, 1)
void hybrid_sampler_kernel(const float* __restrict__ inp,
                           const float* __restrict__ W1,
                           const float* __restrict__ b1,
                           const float* __restrict__ W2,
                           const float* __restrict__ b2,
                           float* __restrict__ out, int nrows) {
  // per-wave private 32x16 f32 tile for D-layout -> A-layout transposes
  __shared__ float hbuf[8][32 * 16];

  const int tid  = threadIdx.x;
  const int wave = tid >> 5;
  const int lane = tid & 31;
  const int n    = lane & 15;        // column index within tile
  const bool hi  = lane >= 16;
  float* hb = hbuf[wave];

  int row = blockIdx.x * 256 + wave * 32 + lane;
  row = (row >= nrows) ? (nrows - 1) : row;   // branchless clamp

  const float in0 = inp[2 * row + 0];
  const float in1 = inp[2 * row + 1];

  // ---- unconditional weight/bias loads (all indices in-bounds), blend after
  const int n8 = n & 7;
  const int n4 = n & 3;
  const float w1k0 = W1[n8];          // W1 row 0
  const float w1k1 = W1[8 + n8];      // W1 row 1
  const float bb1  = b1[n8];
  const int rlo = hi ? 8  : 0;        // W2 K-row (2|0)*4
  const int rl1 = hi ? 12 : 4;        //          (3|1)*4
  const int rh0 = hi ? 24 : 16;       //          (6|4)*4
  const int rh1 = hi ? 28 : 20;       //          (7|5)*4
  const float w2a = W2[rlo + n4];
  const float w2b = W2[rl1 + n4];
  const float w2c = W2[rh0 + n4];
  const float w2d = W2[rh1 + n4];
  const float bb2 = b2[n4];

  // ---------------- layer 1: h_pre = in @ W1 + b1 (K=2 padded to 4) --------
  const float selLo8 = (!hi && n < 8) ? 1.0f : 0.0f;  // B valid only lanes 0..7
  const float m8     = (n < 8) ? 1.0f : 0.0f;
  v2f B0;                                   // B 4x16: v0={K0|K2}, v1={K1|K3}
  B0.x = w1k0 * selLo8;
  B0.y = w1k1 * selLo8;
  const float c1 = bb1 * m8;
  v8f C1 = {c1, c1, c1, c1, c1, c1, c1, c1};

  // A tile 0 = rows 0..15 of this wave (natural); tile 1 = rows 16..31
  v2f A0; A0.x = hi ? 0.0f : in0; A0.y = hi ? 0.0f : in1;
  const int sidx = ((lane + 16) & 31) << 2;
  const float in0s = __int_as_float(__builtin_amdgcn_ds_bpermute(sidx, __float_as_int(in0)));
  const float in1s = __int_as_float(__builtin_amdgcn_ds_bpermute(sidx, __float_as_int(in1)));
  v2f A1; A1.x = hi ? 0.0f : in0s; A1.y = hi ? 0.0f : in1s;

  v8f H0 = __builtin_amdgcn_wmma_f32_16x16x4_f32(false, A0, false, B0, (short)0, C1, false, false);
  v8f H1 = __builtin_amdgcn_wmma_f32_16x16x4_f32(false, A1, false, B0, (short)0, C1, false, false);

  // scatter RAW h_pre (D-layout) -> LDS row-major; tanh is applied after the
  // transpose on A-layout operands (8 values/lane instead of 16 -> no padding waste)
  #pragma unroll
  for (int j = 0; j < 8; ++j) {
    const int m = j + (hi ? 8 : 0);
    hb[m * 16 + n]        = H0[j];
    hb[(m + 16) * 16 + n] = H1[j];
  }

  // ---------------- layer 2: w = tanh(h_pre) @ W2 + b2 ----------------------
  // gather A-layout operands from LDS (DS ops are in-order within a wave)
  const int koff = hi ? 2 : 0;
  v2f AL0 = *(const v2f*)&hb[n * 16 + koff];            // rows 0..15, cols 0..3
  v2f AH0 = *(const v2f*)&hb[n * 16 + 4 + koff];        // rows 0..15, cols 4..7
  v2f AL1 = *(const v2f*)&hb[(n + 16) * 16 + koff];     // rows 16..31
  v2f AH1 = *(const v2f*)&hb[(n + 16) * 16 + 4 + koff];
  AL0.x = fast_tanh(AL0.x);  AL0.y = fast_tanh(AL0.y);
  AH0.x = fast_tanh(AH0.x);  AH0.y = fast_tanh(AH0.y);
  AL1.x = fast_tanh(AL1.x);  AL1.y = fast_tanh(AL1.y);
  AH1.x = fast_tanh(AH1.x);  AH1.y = fast_tanh(AH1.y);

  const float m4 = (n < 4) ? 1.0f : 0.0f;
  v2f BL, BH;
  BL.x = w2a * m4;
  BL.y = w2b * m4;
  BH.x = w2c * m4;
  BH.y = w2d * m4;
  const float c2 = bb2 * m4;
  v8f C2 = {c2, c2, c2, c2, c2, c2, c2, c2};

  v8f T0  = __builtin_amdgcn_wmma_f32_16x16x4_f32(false, AH0, false, BH, (short)0, C2, false, false);
  v8f Wt0 = __builtin_amdgcn_wmma_f32_16x16x4_f32(false, AL0, false, BL, (short)0, T0, false, false);
  v8f T1  = __builtin_amdgcn_wmma_f32_16x16x4_f32(false, AH1, false, BH, (short)0, C2, false, false);
  v8f Wt1 = __builtin_amdgcn_wmma_f32_16x16x4_f32(false, AL1, false, BL, (short)0, T1, false, false);

  // scatter w (D-layout) back to LDS so each lane can read its own row
  #pragma unroll
  for (int j = 0; j < 8; ++j) {
    const int m = j + (hi ? 8 : 0);
    hb[m * 16 + n]        = Wt0[j];
    hb[(m + 16) * 16 + n] = Wt1[j];
  }
  const float4 wv = *(const float4*)&hb[lane * 16];  // w0..w3 for my row

  // ---------------- 2-qubit circuit (state s[2*q0+q1]) ----------------------
  float s0, s1, s2, s3;
  { SCHALF(in0, ss0, cc0); s0 = cc0; s2 = ss0; s1 = 0.0f; s3 = 0.0f; }
  RY_Q1(in1);
  { const float t = s2; s2 = s3; s3 = t; }   // CX: flat perm [0,1,3,2]
  RY_Q0(wv.x);
  RY_Q1(wv.y);
  { const float t = s2; s2 = s3; s3 = t; }   // CX
  RY_Q0(wv.z);
  RY_Q1(wv.w);

  const float p0 = s0 * s0 + s2 * s2;  // P(q1=0)
  const float p1 = s1 * s1 + s3 * s3;  // P(q1=1)
  const float mx = fmaxf(p0, p1);
  const float e0 = __expf(p0 - mx);
  const float e1 = __expf(p1 - mx);
  const float inv = __builtin_amdgcn_rcpf(e0 + e1);
  float2 o; o.x = e0 * inv; o.y = e1 * inv;
  *(float2*)&out[2 * row] = o;         // coalesced B64 store
}

extern "C" void kernel_launch(void* const* d_in, const int* in_sizes, int n_in,
                              void* d_out, int out_size, void* d_ws, size_t ws_size,
                              hipStream_t stream) {
  const float* inp = (const float*)d_in[0];
  const float* W1  = (const float*)d_in[1];
  const float* b1  = (const float*)d_in[2];
  const float* W2  = (const float*)d_in[3];
  const float* b2  = (const float*)d_in[4];
  float* out = (float*)d_out;

  const int nrows  = in_sizes[0] / 2;          // B = 2,097,152
  const int blocks = (nrows + 255) / 256;      // 256 thr = 8 wave32, 32 rows/wave

  hipLaunchKernelGGL(hybrid_sampler_kernel, dim3(blocks), dim3(256), 0, stream,
                     inp, W1, b1, W2, b2, out, nrows);
}